// DendriticLayerSiLU_FFN_87471303950864
// MI455X (gfx1250) — compile-verified
//
#include <hip/hip_runtime.h>
#include <hip/hip_bf16.h>
#include <math.h>

typedef __attribute__((ext_vector_type(2))) float v2f;
typedef __attribute__((ext_vector_type(8))) float v8f;

#define IN_F 1024
#define OUT_F 2048
#define NWIN 16
#define WSZ 64

// One wave computes a 16(token) x 64(feature) output tile: 4 WMMA col-tiles.
// Per window w (16 total): 16 chained V_WMMA_F32_16X16X4_F32 per col-tile
// (K=64), then the softmax/SiLU gating is accumulated online in registers
// (den = sum exp(s), num = sum exp(s)*silu(s)) so the (N,H,16) score tensor
// is never materialized.
__global__ __launch_bounds__(256)
void dendritic_wmma_f32(const float* __restrict__ x,
                        const float* __restrict__ tmpl,
                        float* __restrict__ out,
                        int mtiles, int total_waves)
{
    const int lane = threadIdx.x & 31;
    const int wave = threadIdx.x >> 5;
    const int gw   = blockIdx.x * 8 + wave;
    if (gw >= total_waves) return;            // wave-uniform: EXEC stays all-1s

    const int mt = gw % mtiles;               // token tile (16 rows)
    const int cg = gw / mtiles;               // column group (64 features)
    const int n0 = mt * 16;
    const int h0 = cg * 64;

    const int r  = lane & 15;                 // row within fragment
    const int kh = lane >> 4;                 // selects K pair {0,1} vs {2,3}

    // A fragment source: x row (n0+r), advancing along K; float2 per step.
    const float* xrow = x + (size_t)(n0 + r) * IN_F + kh * 2;
    // B fragment sources: template rows (h0 + j*16 + r) — B[k][n] = tmpl[h][k].
    const float* trow[4];
#pragma unroll
    for (int j = 0; j < 4; ++j)
        trow[j] = tmpl + (size_t)(h0 + j * 16 + r) * IN_F + kh * 2;

    v8f den[4], num[4];
#pragma unroll
    for (int j = 0; j < 4; ++j) {
        den[j] = (v8f){0.f,0.f,0.f,0.f,0.f,0.f,0.f,0.f};
        num[j] = (v8f){0.f,0.f,0.f,0.f,0.f,0.f,0.f,0.f};
    }

    for (int w = 0; w < NWIN; ++w) {
        const int base = w * WSZ;

        v8f acc[4];
#pragma unroll
        for (int j = 0; j < 4; ++j)
            acc[j] = (v8f){0.f,0.f,0.f,0.f,0.f,0.f,0.f,0.f};

#pragma unroll
        for (int k0 = 0; k0 < WSZ; k0 += 4) {
            v2f a = *(const v2f*)(xrow + base + k0);
#pragma unroll
            for (int j = 0; j < 4; ++j) {
                v2f b = *(const v2f*)(trow[j] + base + k0);
                acc[j] = __builtin_amdgcn_wmma_f32_16x16x4_f32(
                    /*neg_a=*/false, a, /*neg_b=*/false, b,
                    /*c_mod=*/(short)0, acc[j],
                    /*reuse_a=*/false, /*reuse_b=*/false);
            }
        }

        // Fused gating epilogue: s ~ N(0,1) so exp(s) never overflows and no
        // running-max is needed. sigmoid(s) = e/(1+e) reuses the softmax exp.
#pragma unroll
        for (int j = 0; j < 4; ++j) {
#pragma unroll
            for (int e = 0; e < 8; ++e) {
                float s  = acc[j][e];
                float ex = __expf(s);
                float sg = ex * __builtin_amdgcn_rcpf(1.0f + ex); // sigmoid(s)
                den[j][e] += ex;
                num[j][e] += ex * (s * sg);                       // exp(s)*silu(s)
            }
        }
    }

    // D layout: VGPR e holds row M=e (lanes 0-15) / M=e+8 (lanes 16-31),
    // column N = lane&15.
#pragma unroll
    for (int j = 0; j < 4; ++j) {
        float* o = out + (size_t)(n0 + kh * 8) * OUT_F + h0 + j * 16 + r;
#pragma unroll
        for (int e = 0; e < 8; ++e) {
            o[(size_t)e * OUT_F] = num[j][e] / den[j][e];
        }
    }
}

extern "C" void kernel_launch(void* const* d_in, const int* in_sizes, int n_in,
                              void* d_out, int out_size, void* d_ws, size_t ws_size,
                              hipStream_t stream) {
    const float* x    = (const float*)d_in[0];
    const float* tmpl = (const float*)d_in[1];
    float* out        = (float*)d_out;

    const int ntok    = in_sizes[0] / IN_F;   // 4096
    const int mtiles  = ntok / 16;            // 256 token tiles
    const int cgroups = OUT_F / 64;           // 32 column groups
    const int waves   = mtiles * cgroups;     // 8192 wave-tiles
    const int blocks  = (waves + 7) / 8;      // 8 waves (256 thr) per block

    dendritic_wmma_f32<<<blocks, 256, 0, stream>>>(x, tmpl, out, mtiles, waves);
}